// MainGNNModel_50689204027567
// MI455X (gfx1250) — compile-verified
//
#include <hip/hip_runtime.h>

#define N_PS 100000
#define N_GW 200000
#define N_SW 100000
#define D_IN 128
#define D_OUT 256
#define NE   400000
#define A_STRIDE 132  // 16-row A stripe in LDS, padded: 132%64=4 banks -> conflict-free, 528B row keeps 16B align

typedef float v2f __attribute__((ext_vector_type(2)));
typedef float v8f __attribute__((ext_vector_type(8)));

// ---------------- utility kernels ----------------

__global__ void zero_kernel(float4* __restrict__ p, long long n4) {
  long long i = (long long)blockIdx.x * blockDim.x + threadIdx.x;
  if (i < n4) p[i] = make_float4(0.f, 0.f, 0.f, 0.f);
}

// One wave (32 lanes) per edge; each lane handles 4 consecutive features.
// Edge index is wave-uniform -> scalar index loads; gathers are b128.
__global__ void scatter_kernel(const float* __restrict__ x,
                               const int* __restrict__ src,
                               const int* __restrict__ dst,
                               float* __restrict__ agg,
                               float* __restrict__ cnt,
                               int nE) {
  int wave_in_blk = threadIdx.x >> 5;
  int q = threadIdx.x & 31;                 // lane within wave
  long long e = (long long)blockIdx.x * (blockDim.x >> 5) + wave_in_blk;
  if (e >= nE) return;
  int s = src[e];
  int t = dst[e];
  int d = q * 4;
  float4 v = *(const float4*)&x[(long long)s * D_IN + d];
  float* ap = &agg[(long long)t * D_IN + d];
  unsafeAtomicAdd(ap + 0, v.x);
  unsafeAtomicAdd(ap + 1, v.y);
  unsafeAtomicAdd(ap + 2, v.z);
  unsafeAtomicAdd(ap + 3, v.w);
  if (q == 0) unsafeAtomicAdd(&cnt[t], 1.0f);
}

// mean = agg / max(cnt,1), vectorized: one float4 per thread
__global__ void normalize_kernel(float4* __restrict__ agg,
                                 const float* __restrict__ cnt,
                                 long long n4) {
  long long idx = (long long)blockIdx.x * blockDim.x + threadIdx.x;
  if (idx >= n4) return;
  float c = fmaxf(cnt[idx >> 5], 1.0f);   // 32 float4 per 128-feature row
  float4 v = agg[idx];
  v.x /= c; v.y /= c; v.z /= c; v.w /= c;
  agg[idx] = v;
}

// Wrc = Wr_a + Wr_b (128x256), bc = b_a + b_b (256)
__global__ void combine_kernel(const float* __restrict__ wa, const float* __restrict__ wb,
                               float* __restrict__ wo,
                               const float* __restrict__ ba, const float* __restrict__ bb,
                               float* __restrict__ bo, int n) {
  int i = blockIdx.x * blockDim.x + threadIdx.x;
  if (i < n) wo[i] = wa[i] + wb[i];
  if (i < D_OUT) bo[i] = ba[i] + bb[i];
}

// ---------------- WMMA GEMM ----------------
// One GEMM term: C(two 16x16 tiles per wave) += A[16 x 128] * B[128 x 32-slice]
// A staged in LDS; fragments per V_WMMA_F32_16X16X4_F32 ISA layout:
//   A 16x4:  lane(0-15)=M, vgpr0/1 = K0/K1 ; lanes 16-31 carry K2/K3
//   B 4x16:  vgpr0 lanes0-15 = row K0, lanes16-31 = row K2 (row-striped)
//   C 16x16: vgpr r -> rows r (lanes0-15) and r+8 (lanes16-31), col = lane%16
__device__ __forceinline__ void mma_term(const float* __restrict__ A,
                                         const float* __restrict__ B,
                                         float* Alds, int r0, int n_rows,
                                         int tid, int wave, int half, int lh,
                                         v8f& c0, v8f& c1) {
  __syncthreads();  // protect previous term's LDS reads
  // stage 16x128 A stripe as float4: 512 vec elements, 2 per thread
  for (int i = tid; i < 16 * (D_IN / 4); i += 256) {
    int rr = i >> 5;            // row in stripe
    int kv = i & 31;            // float4 index within row
    int gr = r0 + rr;
    if (gr >= n_rows) gr = n_rows - 1;  // clamped dummy rows (masked at store)
    float4 v = *(const float4*)&A[(long long)gr * D_IN + kv * 4];
    *(float4*)&Alds[rr * A_STRIDE + kv * 4] = v;
  }
  __syncthreads();

  const int col0 = wave * 32 + lh;
  const int col1 = col0 + 16;
#pragma unroll 8
  for (int k0 = 0; k0 < D_IN; k0 += 4) {
    int ka = k0 + 2 * half;
    v2f a;
    a.x = Alds[lh * A_STRIDE + ka];
    a.y = Alds[lh * A_STRIDE + ka + 1];
    const float* Brow0 = B + (long long)ka * D_OUT;
    const float* Brow1 = B + (long long)(ka + 1) * D_OUT;
    v2f b0, b1;
    b0.x = Brow0[col0];
    b0.y = Brow1[col0];
    b1.x = Brow0[col1];
    b1.y = Brow1[col1];
    c0 = __builtin_amdgcn_wmma_f32_16x16x4_f32(false, a, false, b0, (short)0, c0, false, false);
    c1 = __builtin_amdgcn_wmma_f32_16x16x4_f32(false, a, false, b1, (short)0, c1, false, false);
  }
}

// out = relu(sum_t A_t @ B_t + bias). HEAD: instead emit prelu(relu(.)@Wlin + blin).
template <bool HEAD>
__launch_bounds__(256)
__global__ void sage_gemm_kernel(const float* __restrict__ A0, const float* __restrict__ B0,
                                 const float* __restrict__ A1, const float* __restrict__ B1,
                                 const float* __restrict__ A2, const float* __restrict__ B2,
                                 const float* __restrict__ bias, int n_rows,
                                 float* __restrict__ out,
                                 const float* __restrict__ Wlin,
                                 const float* __restrict__ blin,
                                 const float* __restrict__ prelu,
                                 float* __restrict__ gw) {
  __shared__ float Alds[16 * A_STRIDE];
  __shared__ float redbuf[8 * 16];

  const int tid  = threadIdx.x;
  const int wave = tid >> 5;
  const int lane = tid & 31;
  const int half = lane >> 4;
  const int lh   = lane & 15;
  const int r0   = blockIdx.x * 16;

  v8f c0 = {0.f, 0.f, 0.f, 0.f, 0.f, 0.f, 0.f, 0.f};
  v8f c1 = {0.f, 0.f, 0.f, 0.f, 0.f, 0.f, 0.f, 0.f};

  mma_term(A0, B0, Alds, r0, n_rows, tid, wave, half, lh, c0, c1);
  if (A1) mma_term(A1, B1, Alds, r0, n_rows, tid, wave, half, lh, c0, c1);
  if (A2) mma_term(A2, B2, Alds, r0, n_rows, tid, wave, half, lh, c0, c1);

  const int col0 = wave * 32 + lh;
  const int col1 = col0 + 16;
  const float bb0 = bias[col0];
  const float bb1 = bias[col1];

  float v0[8], v1[8];
#pragma unroll
  for (int r = 0; r < 8; ++r) {
    v0[r] = fmaxf(c0[r] + bb0, 0.0f);
    v1[r] = fmaxf(c1[r] + bb1, 0.0f);
  }

  if (!HEAD) {
#pragma unroll
    for (int r = 0; r < 8; ++r) {
      int row = r0 + r + 8 * half;
      if (row < n_rows) {
        out[(long long)row * D_OUT + col0] = v0[r];
        out[(long long)row * D_OUT + col1] = v1[r];
      }
    }
  } else {
    // per-row dot with Wlin over this wave's 32 columns
    const float w0 = Wlin[col0];
    const float w1 = Wlin[col1];
    float p[8];
#pragma unroll
    for (int r = 0; r < 8; ++r) p[r] = v0[r] * w0 + v1[r] * w1;
    // reduce within each 16-lane half (xor masks < 16 stay inside the half)
#pragma unroll
    for (int m = 1; m < 16; m <<= 1) {
#pragma unroll
      for (int r = 0; r < 8; ++r) p[r] += __shfl_xor(p[r], m, 32);
    }
    if (lh == 0) {
#pragma unroll
      for (int r = 0; r < 8; ++r) redbuf[wave * 16 + r + 8 * half] = p[r];
    }
    __syncthreads();
    if (tid < 16) {
      float s = 0.0f;
#pragma unroll
      for (int w = 0; w < 8; ++w) s += redbuf[w * 16 + tid];
      s += blin[0];
      s = (s >= 0.0f) ? s : prelu[0] * s;
      int row = r0 + tid;
      if (row < n_rows) gw[row] = s;
    }
  }
}

// ---------------- launch ----------------

extern "C" void kernel_launch(void* const* d_in, const int* in_sizes, int n_in,
                              void* d_out, int out_size, void* d_ws, size_t ws_size,
                              hipStream_t stream) {
  const float* x_ps = (const float*)d_in[0];
  const float* x_gw = (const float*)d_in[1];
  const float* x_sw = (const float*)d_in[2];
  const int* src_ps_gw = (const int*)d_in[3];
  const int* dst_ps_gw = (const int*)d_in[4];
  const int* src_gw_ps = (const int*)d_in[5];
  const int* dst_gw_ps = (const int*)d_in[6];
  const int* src_ps_sw = (const int*)d_in[7];
  const int* dst_ps_sw = (const int*)d_in[8];
  const int* src_sw_ps = (const int*)d_in[9];
  const int* dst_sw_ps = (const int*)d_in[10];
  const float* Wl_ps_gw = (const float*)d_in[11];
  const float* Wr_ps_gw = (const float*)d_in[12];
  const float* b_ps_gw  = (const float*)d_in[13];
  const float* Wl_gw_ps = (const float*)d_in[14];
  const float* Wr_gw_ps = (const float*)d_in[15];
  const float* b_gw_ps  = (const float*)d_in[16];
  const float* Wl_ps_sw = (const float*)d_in[17];
  const float* Wr_ps_sw = (const float*)d_in[18];
  const float* b_ps_sw  = (const float*)d_in[19];
  const float* Wl_sw_ps = (const float*)d_in[20];
  const float* Wr_sw_ps = (const float*)d_in[21];
  const float* b_sw_ps  = (const float*)d_in[22];
  const float* W_lin    = (const float*)d_in[23];
  const float* b_lin    = (const float*)d_in[24];
  const float* prelu_w  = (const float*)d_in[25];

  // workspace layout (floats)
  float* ws = (float*)d_ws;
  float* agg_gw  = ws;                                   // 200000*128
  float* agg_ps1 = agg_gw  + (long long)N_GW * D_IN;     // 100000*128 (from gw)
  float* agg_ps2 = agg_ps1 + (long long)N_PS * D_IN;     // 100000*128 (from sw)
  float* agg_sw  = agg_ps2 + (long long)N_PS * D_IN;     // 100000*128
  float* cnt_gw  = agg_sw  + (long long)N_SW * D_IN;     // 200000
  float* cnt_ps1 = cnt_gw  + N_GW;                       // 100000
  float* cnt_ps2 = cnt_ps1 + N_PS;                       // 100000
  float* cnt_sw  = cnt_ps2 + N_PS;                       // 100000
  float* Wrc     = cnt_sw  + N_SW;                       // 128*256
  float* bc      = Wrc + D_IN * D_OUT;                   // 256

  // outputs (concatenated: gw [N_GW], out_ps [N_PS*256], out_sw [N_SW*256])
  float* out    = (float*)d_out;
  float* gw_out = out;
  float* out_ps = out + N_GW;
  float* out_sw = out_ps + (long long)N_PS * D_OUT;

  const int TB = 256;

  // 1) zero agg + cnt buffers (all sizes multiple of 4 floats)
  long long nz  = (long long)(N_GW + 2 * N_PS + N_SW) * D_IN + (N_GW + 2 * N_PS + N_SW);
  long long nz4 = nz / 4;
  zero_kernel<<<(int)((nz4 + TB - 1) / TB), TB, 0, stream>>>((float4*)ws, nz4);

  // 2) scatter-add per relation: one wave per edge (32 lanes x float4)
  int waves_per_blk = TB / 32;  // 8
  int sblocks = (NE + waves_per_blk - 1) / waves_per_blk;
  scatter_kernel<<<sblocks, TB, 0, stream>>>(x_ps, src_ps_gw, dst_ps_gw, agg_gw,  cnt_gw,  NE);
  scatter_kernel<<<sblocks, TB, 0, stream>>>(x_gw, src_gw_ps, dst_gw_ps, agg_ps1, cnt_ps1, NE);
  scatter_kernel<<<sblocks, TB, 0, stream>>>(x_sw, src_sw_ps, dst_sw_ps, agg_ps2, cnt_ps2, NE);
  scatter_kernel<<<sblocks, TB, 0, stream>>>(x_ps, src_ps_sw, dst_ps_sw, agg_sw,  cnt_sw,  NE);

  // 3) mean = agg / max(cnt,1)  (float4 per thread; 32 vec per row)
  long long ngw4 = (long long)N_GW * D_IN / 4;
  long long nps4 = (long long)N_PS * D_IN / 4;
  long long nsw4 = (long long)N_SW * D_IN / 4;
  normalize_kernel<<<(int)((ngw4 + TB - 1) / TB), TB, 0, stream>>>((float4*)agg_gw,  cnt_gw,  ngw4);
  normalize_kernel<<<(int)((nps4 + TB - 1) / TB), TB, 0, stream>>>((float4*)agg_ps1, cnt_ps1, nps4);
  normalize_kernel<<<(int)((nps4 + TB - 1) / TB), TB, 0, stream>>>((float4*)agg_ps2, cnt_ps2, nps4);
  normalize_kernel<<<(int)((nsw4 + TB - 1) / TB), TB, 0, stream>>>((float4*)agg_sw,  cnt_sw,  nsw4);

  // 4) combine the two root weights/biases feeding ps
  combine_kernel<<<(D_IN * D_OUT + TB - 1) / TB, TB, 0, stream>>>(
      Wr_gw_ps, Wr_sw_ps, Wrc, b_gw_ps, b_sw_ps, bc, D_IN * D_OUT);

  // 5) WMMA GEMMs
  // out_ps = relu(mean1@Wl_gw_ps + mean2@Wl_sw_ps + x_ps@(Wr_gw_ps+Wr_sw_ps) + (b1+b2))
  sage_gemm_kernel<false><<<(N_PS + 15) / 16, TB, 0, stream>>>(
      agg_ps1, Wl_gw_ps, agg_ps2, Wl_sw_ps, x_ps, Wrc, bc, N_PS, out_ps,
      nullptr, nullptr, nullptr, nullptr);
  // out_sw = relu(mean_sw@Wl_ps_sw + x_sw@Wr_ps_sw + b)
  sage_gemm_kernel<false><<<(N_SW + 15) / 16, TB, 0, stream>>>(
      agg_sw, Wl_ps_sw, x_sw, Wr_ps_sw, nullptr, nullptr, b_ps_sw, N_SW, out_sw,
      nullptr, nullptr, nullptr, nullptr);
  // gw = prelu(relu(mean_gw@Wl_ps_gw + x_gw@Wr_ps_gw + b) @ W_lin + b_lin)  [fused head]
  sage_gemm_kernel<true><<<(N_GW + 15) / 16, TB, 0, stream>>>(
      agg_gw, Wl_ps_gw, x_gw, Wr_ps_gw, nullptr, nullptr, b_ps_gw, N_GW, nullptr,
      W_lin, b_lin, prelu_w, gw_out);
}